// trend_stgcn_76424648065380
// MI455X (gfx1250) — compile-verified
//
#include <hip/hip_runtime.h>
#include <math.h>

typedef __attribute__((ext_vector_type(2))) float v2f;
typedef __attribute__((ext_vector_type(8))) float v8f;

#define Bn   128
#define Nn   64
#define Tt   512
#define Kk   5
#define NF   257           // rfft bins
#define NCOL 528           // 2*257=514 padded to 33*16
#define ROWS (Bn * Nn)     // 8192
#define PADE 1092          // (pmax-1)//2 + 1, pmax = 512*512//120 = 2184
#define PMAX 2184

// ---------------- Kernel 1: DFT matrix, column-major W2[col][t] ---------------
// col 2f   -> cos(2*pi*t*f/512)
// col 2f+1 -> -sin(2*pi*t*f/512)
// cols 514..527 -> 0 (padding so GEMM tiles are uniform)
__global__ void gen_w_kernel(float* __restrict__ W2) {
    int i = blockIdx.x * blockDim.x + threadIdx.x;
    if (i >= NCOL * Tt) return;
    int col = i >> 9;          // / 512
    int t   = i & 511;
    float v = 0.0f;
    if (col < 2 * NF) {
        int f = col >> 1;
        int m = (t * f) & 511; // (t*f) mod 512 keeps the angle in [0, 2*pi)
        float theta = (float)m * (6.28318530717958647692f / 512.0f);
        v = (col & 1) ? -sinf(theta) : cosf(theta);
    }
    W2[i] = v;
}

// ---------------- Kernel 2: DFT as GEMM via fp32 WMMA ------------------------
// C[8192 x 528] = x[8192 x 512] * W[512 x 528]; one 16x16 tile per wave,
// K stepped by 4 with V_WMMA_F32_16X16X4_F32 (128 wmma ops per wave).
// A frag (ISA 7.12.2, 32-bit A 16x4): lane l holds (M=l&15, K=2*(l>>4)+v).
// B frag mirrors A with N=l&15; W2 is column-major so both are float2 loads.
__global__ void dft_wmma_kernel(const float* __restrict__ x,
                                const float* __restrict__ W2,
                                float* __restrict__ RI) {
    int lane  = threadIdx.x & 31;
    int g     = blockIdx.x * (blockDim.x >> 5) + (threadIdx.x >> 5);
    int mTile = g & 511;       // 512 M-tiles
    int nTile = g >> 9;        // 33 N-tiles
    int half  = lane >> 4;
    int sub   = lane & 15;

    const float* arow = x  + (size_t)(mTile * 16 + sub) * Tt + 2 * half;
    const float* brow = W2 + (size_t)(nTile * 16 + sub) * Tt + 2 * half;

    v8f acc = {0.f, 0.f, 0.f, 0.f, 0.f, 0.f, 0.f, 0.f};
#pragma unroll 4
    for (int kk = 0; kk < Tt; kk += 4) {
        v2f a = *(const v2f*)(arow + kk);
        v2f b = *(const v2f*)(brow + kk);
        acc = __builtin_amdgcn_wmma_f32_16x16x4_f32(
            /*neg_a=*/false, a, /*neg_b=*/false, b,
            /*c_mod=*/(short)0, acc, /*reuse_a=*/false, /*reuse_b=*/false);
    }

    // C/D layout: lane l, vgpr v -> (M = 8*(l>>4)+v, N = l&15)
    int row0 = mTile * 16 + 8 * half;
    int col  = nTile * 16 + sub;
    float* out = RI + (size_t)row0 * NCOL + col;
#pragma unroll
    for (int v = 0; v < 8; ++v) out[(size_t)v * NCOL] = acc[v];
}

// ---------------- Kernel 3: magnitude, mean over channels --------------------
__global__ void mag_kernel(const float* __restrict__ RI, float* __restrict__ mag) {
    int idx = blockIdx.x * blockDim.x + threadIdx.x;
    if (idx >= Bn * NF) return;
    int b = idx / NF, f = idx % NF;
    if (f == 0) { mag[idx] = 0.0f; return; }   // mag[:,0] = 0
    const float* base = RI + (size_t)b * Nn * NCOL + 2 * f;
    float s = 0.0f;
    for (int n = 0; n < Nn; ++n) {
        float re = base[(size_t)n * NCOL];
        float im = base[(size_t)n * NCOL + 1];
        s += sqrtf(re * re + im * im);
    }
    mag[idx] = s * (1.0f / (float)Nn);
}

// ---------------- Kernel 4: top-5 per batch -> window sizes ------------------
// Strict '>' keeps the lowest index on ties (matches jax.lax.top_k).
// freqs[i] = i * (120/512); 120/512 = 0.234375 is exact in fp32, so the
// float divide + floor matches the reference arithmetic.
__global__ void topk_kernel(const float* __restrict__ mag, int* __restrict__ scaleArr) {
    int b = blockIdx.x * blockDim.x + threadIdx.x;
    if (b >= Bn) return;
    const float* m = mag + (size_t)b * NF;
    int ch[Kk];
    for (int j = 0; j < Kk; ++j) {
        float best = -1.0f; int bi = 0;
        for (int f = 0; f < NF; ++f) {
            bool used = false;
            for (int q = 0; q < j; ++q) used |= (ch[q] == f);
            float v = m[f];
            if (!used && v > best) { best = v; bi = f; }
        }
        ch[j] = bi;
        int k;
        if (bi == 0) {
            k = 1;  // cannot happen (mag[0]==0 < all others), defensive
        } else {
            float frf = (float)bi * (120.0f / 512.0f);
            k = (int)floorf(512.0f / frf);
        }
        k = min(max(k, 1), PMAX);
        scaleArr[b * Kk + j] = k;
    }
}

// ---------------- Kernel 5: per-row prefix sums C[513] -----------------------
// One wave32 per (b,n) row: 16 contiguous elements per lane, shuffle scan.
__global__ void cumsum_kernel(const float* __restrict__ x, float* __restrict__ C) {
    int row  = blockIdx.x * (blockDim.x >> 5) + (threadIdx.x >> 5);
    int lane = threadIdx.x & 31;
    const float* xr = x + (size_t)row * Tt;
    float v[16];
    float s = 0.0f;
#pragma unroll
    for (int i = 0; i < 16; ++i) { s += xr[lane * 16 + i]; v[i] = s; }
    float t = s;
    for (int d = 1; d < 32; d <<= 1) {
        float u = __shfl_up(t, d, 32);
        if (lane >= d) t += u;
    }
    float base = t - s;  // exclusive prefix of lane totals
    float* Cr = C + (size_t)row * (Tt + 1);
    if (lane == 0) Cr[0] = 0.0f;
#pragma unroll
    for (int i = 0; i < 16; ++i) Cr[1 + lane * 16 + i] = base + v[i];
}

// ---------------- Kernel 6: windowed means ----------------------------------
// S(m) of the edge-clamped extension, analytically:
//   m <= pad        : m*x0
//   pad..pad+512    : pad*x0 + C[m-pad]
//   m >= pad+512    : pad*x0 + C[512] + (m-pad-512)*x_last
__device__ __forceinline__ float Sfun(int m, float x0, const float* sC,
                                      float tot, float xl) {
    if (m <= PADE) return (float)m * x0;
    if (m <= PADE + Tt) return (float)PADE * x0 + sC[m - PADE];
    return (float)PADE * x0 + tot + (float)(m - PADE - Tt) * xl;
}

__global__ void windows_kernel(const float* __restrict__ C,
                               const int* __restrict__ scaleArr,
                               float* __restrict__ out) {
    int bn = blockIdx.x;          // 0..8191  (b*64 + n)
    int b  = bn >> 6;
    int t  = threadIdx.x;         // 0..511
    __shared__ float sC[Tt + 1];
    const float* Cr = C + (size_t)bn * (Tt + 1);
    for (int i = t; i < Tt + 1; i += Tt) sC[i] = Cr[i];
    __syncthreads();
    float x0  = sC[1];
    float tot = sC[Tt];
    float xl  = sC[Tt] - sC[Tt - 1];
    for (int kidx = 0; kidx < Kk; ++kidx) {
        int k = scaleArr[b * Kk + kidx];
        int p = (k - 1) >> 1;
        int Lm1 = Tt + 2 * p - k;         // L - 1
        int tc = min(t, Lm1);
        int start = tc - p + PADE;
        float hi = Sfun(start + k, x0, sC, tot, xl);
        float lo = Sfun(start,     x0, sC, tot, xl);
        out[(((size_t)bn * Kk + kidx) << 9) + t] = (hi - lo) / (float)k;
    }
}

// ---------------- launcher ---------------------------------------------------
extern "C" void kernel_launch(void* const* d_in, const int* in_sizes, int n_in,
                              void* d_out, int out_size, void* d_ws, size_t ws_size,
                              hipStream_t stream) {
    const float* x = (const float*)d_in[0];
    float* out = (float*)d_out;

    // scratch layout (~35.3 MB)
    float* W2  = (float*)d_ws;                 // 528*512
    float* RI  = W2 + (size_t)NCOL * Tt;       // 8192*528
    float* mag = RI + (size_t)ROWS * NCOL;     // 128*257
    int*   scl = (int*)(mag + (size_t)Bn * NF);// 128*5
    float* C   = (float*)(scl + Bn * Kk);      // 8192*513

    gen_w_kernel<<<(NCOL * Tt + 255) / 256, 256, 0, stream>>>(W2);

    // 512 M-tiles * 33 N-tiles = 16896 waves, 8 waves/block
    dft_wmma_kernel<<<16896 / 8, 256, 0, stream>>>(x, W2, RI);

    mag_kernel<<<(Bn * NF + 255) / 256, 256, 0, stream>>>(RI, mag);

    topk_kernel<<<1, 128, 0, stream>>>(mag, scl);

    cumsum_kernel<<<ROWS / 8, 256, 0, stream>>>(x, C);

    windows_kernel<<<ROWS, Tt, 0, stream>>>(C, scl, out);
}